// GConv_15118284882196
// MI455X (gfx1250) — compile-verified
//
#include <hip/hip_runtime.h>
#include <stdint.h>

// Problem constants (match reference)
#define NN 50000
#define NE 640000
#define NG 256
#define HD 128
#define NL 5
#define AF 9
#define AV 64
#define BF 3
#define BV 8
#define BN_EPS 1e-5f

typedef float v2f __attribute__((ext_vector_type(2)));
typedef float v8f __attribute__((ext_vector_type(8)));

// ---------------------------------------------------------------------------
// Atom encoder: z[n,:] = sum_f atom_emb[f, x[n,f], :]
// One wave per node, lane handles 4 floats. atom_emb (288 KB) is L2-resident.
// ---------------------------------------------------------------------------
__global__ __launch_bounds__(256)
void atom_encoder_kernel(const int* __restrict__ x,
                         const float* __restrict__ atom_emb,
                         float* __restrict__ z) {
  int tid  = blockIdx.x * 256 + threadIdx.x;
  int n    = tid >> 5;
  int lane = tid & 31;
  if (n >= NN) return;
  int c = lane * 4;
  float4 acc = make_float4(0.f, 0.f, 0.f, 0.f);
#pragma unroll
  for (int f = 0; f < AF; ++f) {
    int a = x[n * AF + f];                       // wave-uniform -> scalar load
    const float4 v =
        *reinterpret_cast<const float4*>(&atom_emb[((f * AV + a) * HD) + c]);
    acc.x += v.x; acc.y += v.y; acc.z += v.z; acc.w += v.w;
  }
  *reinterpret_cast<float4*>(&z[n * HD + c]) = acc;
}

// ---------------------------------------------------------------------------
// aggr = z  (self term of GINEConv eps=0, so GEMM1 input (z+aggr) == aggr)
// ---------------------------------------------------------------------------
__global__ __launch_bounds__(256)
void copy_z_kernel(const float* __restrict__ z, float* __restrict__ aggr) {
  int tid = blockIdx.x * 256 + threadIdx.x;          // NN*HD/4 float4s
  if (tid >= NN * HD / 4) return;
  reinterpret_cast<float4*>(aggr)[tid] =
      reinterpret_cast<const float4*>(z)[tid];
}

// ---------------------------------------------------------------------------
// Edge messages: aggr[dst] += relu(z[src] + bond_enc(edge_attr))
// bond_emb (12 KB) regathered per edge -> the 327 MB/layer e tensor is never
// materialized. z/aggr (25.6 MB each) are L2-resident (192 MB L2), so the
// random gather and the f32 scatter atomics stay on-chip.
// ---------------------------------------------------------------------------
__global__ __launch_bounds__(256)
void edge_msg_kernel(const int* __restrict__ edge_index,
                     const int* __restrict__ edge_attr,
                     const float* __restrict__ bond_emb,
                     const float* __restrict__ z,
                     float* __restrict__ aggr) {
  int e    = blockIdx.x * 8 + (threadIdx.x >> 5);
  int lane = threadIdx.x & 31;
  if (e >= NE) return;
  int src = edge_index[e];                       // wave-uniform
  int dst = edge_index[NE + e];
  int a0  = edge_attr[e * BF + 0];
  int a1  = edge_attr[e * BF + 1];
  int a2  = edge_attr[e * BF + 2];
  int c = lane * 4;
  const float4 e0 =
      *reinterpret_cast<const float4*>(&bond_emb[(0 * BV + a0) * HD + c]);
  const float4 e1 =
      *reinterpret_cast<const float4*>(&bond_emb[(1 * BV + a1) * HD + c]);
  const float4 e2 =
      *reinterpret_cast<const float4*>(&bond_emb[(2 * BV + a2) * HD + c]);
  const float4 zv = *reinterpret_cast<const float4*>(&z[src * HD + c]);
  float m0 = fmaxf(zv.x + e0.x + e1.x + e2.x, 0.f);
  float m1 = fmaxf(zv.y + e0.y + e1.y + e2.y, 0.f);
  float m2 = fmaxf(zv.z + e0.z + e1.z + e2.z, 0.f);
  float m3 = fmaxf(zv.w + e0.w + e1.w + e2.w, 0.f);
  float* out = &aggr[dst * HD + c];
  atomicAdd(out + 0, m0);
  atomicAdd(out + 1, m1);
  atomicAdd(out + 2, m2);
  atomicAdd(out + 3, m3);
}

// ---------------------------------------------------------------------------
// Fused per-layer node MLP, full fp32 via V_WMMA_F32_16X16X4_F32:
//   h  = relu(BN1(A @ W1 + b1))     A = aggr tile [16,128], W1 [128,256]
//   z' = BN2(h @ W2 + b2) (+relu)   h kept in LDS [16,256], W2 [256,128]
// Block = 128 threads = 4 waves, one 16-row M-tile per block (NN % 16 == 0,
// so no row guards anywhere -> EXEC all-ones around WMMA as the ISA requires).
// A tile staged with GLOBAL_LOAD_ASYNC_TO_LDS_B128 (ASYNCcnt path): memory ->
// LDS with no VGPR round-trip, completed with s_wait_asynccnt before barrier.
// LDS rows padded (132 / 260 floats) so 16-lane A-frag reads hit distinct
// banks (lane*132 % 64 = lane*4). Fragment layouts per ISA 7.12.2 (32-bit):
//   lo=lane&15, hi=lane>>4,
//   a[v] = A[lo][kk+2*hi+v], b[v] = W[kk+2*hi+v][nn+lo], d[r] = D[r+8*hi][lo].
// ---------------------------------------------------------------------------
__global__ __launch_bounds__(128)
void gine_mlp_kernel(const float* __restrict__ A,    // aggr [NN,HD]
                     const float* __restrict__ W1,   // [HD, 2*HD] layer slice
                     const float* __restrict__ b1,
                     const float* __restrict__ g1,
                     const float* __restrict__ be1,
                     const float* __restrict__ m1,
                     const float* __restrict__ v1,
                     const float* __restrict__ W2,   // [2*HD, HD]
                     const float* __restrict__ b2,
                     const float* __restrict__ g2,
                     const float* __restrict__ be2,
                     const float* __restrict__ m2,
                     const float* __restrict__ v2,
                     float* __restrict__ zout,       // [NN,HD] next-layer z
                     float* __restrict__ zcat,       // [NN, NL*HD] output slab
                     int layer, int do_relu) {
  __shared__ float As[16 * 132];   // A tile, padded (row = 528 B, 16-aligned)
  __shared__ float Hs[16 * 260];   // GEMM1 out / GEMM2 in, padded

  const int row0 = blockIdx.x * 16;
  const int tid  = threadIdx.x;
  const int wave = tid >> 5;
  const int lane = tid & 31;
  const int lo   = lane & 15;
  const int hi   = lane >> 4;

  // ---- Async global->LDS staging of A tile [16 x 128] (4 x b128/thread) ----
  {
    int i = tid;
#pragma unroll
    for (int it = 0; it < 4; ++it, i += 128) {
      int r  = i >> 5;                          // row in tile
      int c4 = (i & 31) * 4;                    // col (float index)
      unsigned ldsa = (unsigned)(uintptr_t)&As[r * 132 + c4];  // addr[31:0]
      unsigned goff = (unsigned)(((row0 + r) * HD + c4) * sizeof(float));
      asm volatile("global_load_async_to_lds_b128 %0, %1, %2"
                   :
                   : "v"(ldsa), "v"(goff), "s"(A)
                   : "memory");
    }
    asm volatile("s_wait_asynccnt 0x0" ::: "memory");
  }
  __syncthreads();

  // ---- GEMM1: [16,128] @ [128,256], fused bias+BN+ReLU into LDS ----
#pragma unroll
  for (int t = 0; t < 4; ++t) {
    const int nn = wave * 64 + t * 16;
    v8f acc = {};
#pragma unroll 4
    for (int kk = 0; kk < HD; kk += 4) {
      v2f a, b;
      a.x = As[lo * 132 + kk + 2 * hi + 0];
      a.y = As[lo * 132 + kk + 2 * hi + 1];
      b.x = W1[(kk + 2 * hi + 0) * (2 * HD) + nn + lo];
      b.y = W1[(kk + 2 * hi + 1) * (2 * HD) + nn + lo];
      acc = __builtin_amdgcn_wmma_f32_16x16x4_f32(false, a, false, b,
                                                  (short)0, acc, false, false);
    }
    const int c = nn + lo;                 // this lane's output column
    const float alpha = g1[c] * rsqrtf(v1[c] + BN_EPS);
    const float beta  = be1[c] + (b1[c] - m1[c]) * alpha;
#pragma unroll
    for (int r = 0; r < 8; ++r) {
      float v = acc[r] * alpha + beta;     // dot*alpha + (b1-m1)*alpha + be1
      Hs[(r + 8 * hi) * 260 + c] = fmaxf(v, 0.f);
    }
  }
  __syncthreads();

  // ---- GEMM2: [16,256] @ [256,128], fused bias+BN(+ReLU), write out ----
#pragma unroll
  for (int t = 0; t < 2; ++t) {
    const int nn = (wave * 2 + t) * 16;
    v8f acc = {};
#pragma unroll 4
    for (int kk = 0; kk < 2 * HD; kk += 4) {
      v2f a, b;
      a.x = Hs[lo * 260 + kk + 2 * hi + 0];
      a.y = Hs[lo * 260 + kk + 2 * hi + 1];
      b.x = W2[(kk + 2 * hi + 0) * HD + nn + lo];
      b.y = W2[(kk + 2 * hi + 1) * HD + nn + lo];
      acc = __builtin_amdgcn_wmma_f32_16x16x4_f32(false, a, false, b,
                                                  (short)0, acc, false, false);
    }
    const int c = nn + lo;
    const float alpha = g2[c] * rsqrtf(v2[c] + BN_EPS);
    const float beta  = be2[c] + (b2[c] - m2[c]) * alpha;
#pragma unroll
    for (int r = 0; r < 8; ++r) {
      const int row = row0 + r + 8 * hi;   // always < NN (NN % 16 == 0)
      float v = acc[r] * alpha + beta;
      if (do_relu) v = fmaxf(v, 0.f);
      zout[row * HD + c] = v;
      zcat[(size_t)row * (NL * HD) + layer * HD + c] = v;
    }
  }
}

// ---------------------------------------------------------------------------
// Graph readout: g_cat[batch[n], :] += z_cat[n, :]  (g_cat is 640 KB -> L2)
// ---------------------------------------------------------------------------
__global__ __launch_bounds__(256)
void zero_kernel(float* __restrict__ p, int n) {
  int tid = blockIdx.x * 256 + threadIdx.x;
  if (tid < n) p[tid] = 0.f;
}

__global__ __launch_bounds__(256)
void pool_kernel(const int* __restrict__ batch,
                 const float* __restrict__ zcat,
                 float* __restrict__ gcat) {
  int tid = blockIdx.x * 256 + threadIdx.x;          // NN * 160 threads
  if (tid >= NN * (NL * HD / 4)) return;
  int n  = tid / (NL * HD / 4);
  int c4 = (tid % (NL * HD / 4)) * 4;
  int b  = batch[n];
  const float4 v =
      *reinterpret_cast<const float4*>(&zcat[(size_t)n * (NL * HD) + c4]);
  float* out = &gcat[b * (NL * HD) + c4];
  atomicAdd(out + 0, v.x);
  atomicAdd(out + 1, v.y);
  atomicAdd(out + 2, v.z);
  atomicAdd(out + 3, v.w);
}

// ---------------------------------------------------------------------------
extern "C" void kernel_launch(void* const* d_in, const int* in_sizes, int n_in,
                              void* d_out, int out_size, void* d_ws,
                              size_t ws_size, hipStream_t stream) {
  const int*   x          = (const int*)d_in[0];
  const int*   edge_index = (const int*)d_in[1];
  const int*   edge_attr  = (const int*)d_in[2];
  const int*   batch      = (const int*)d_in[3];
  const float* atom_emb   = (const float*)d_in[4];
  const float* bond_emb   = (const float*)d_in[5];
  const float* W1 = (const float*)d_in[6];
  const float* b1 = (const float*)d_in[7];
  const float* g1 = (const float*)d_in[8];
  const float* be1 = (const float*)d_in[9];
  const float* m1 = (const float*)d_in[10];
  const float* v1 = (const float*)d_in[11];
  const float* W2 = (const float*)d_in[12];
  const float* b2 = (const float*)d_in[13];
  const float* g2 = (const float*)d_in[14];
  const float* be2 = (const float*)d_in[15];
  const float* m2 = (const float*)d_in[16];
  const float* v2 = (const float*)d_in[17];

  float* zcat = (float*)d_out;                                  // [NN, 640]
  float* gcat = (float*)d_out + (size_t)NN * (NL * HD);         // [NG, 640]

  float* z    = (float*)d_ws;                                   // [NN, HD]
  float* aggr = z + (size_t)NN * HD;                            // [NN, HD]

  // Atom encoder
  atom_encoder_kernel<<<(NN * 32 + 255) / 256, 256, 0, stream>>>(x, atom_emb, z);

  for (int l = 0; l < NL; ++l) {
    // aggr = z
    copy_z_kernel<<<(NN * HD / 4 + 255) / 256, 256, 0, stream>>>(z, aggr);
    // aggr[dst] += relu(z[src] + bond_enc(edge_attr))
    edge_msg_kernel<<<(NE + 7) / 8, 256, 0, stream>>>(edge_index, edge_attr,
                                                      bond_emb, z, aggr);
    // Fused MLP + BN + ReLU, writes z (next layer) and z_cat slice
    gine_mlp_kernel<<<NN / 16, 128, 0, stream>>>(
        aggr,
        W1 + (size_t)l * HD * 2 * HD, b1 + l * 2 * HD, g1 + l * 2 * HD,
        be1 + l * 2 * HD, m1 + l * 2 * HD, v1 + l * 2 * HD,
        W2 + (size_t)l * 2 * HD * HD, b2 + l * HD, g2 + l * HD,
        be2 + l * HD, m2 + l * HD, v2 + l * HD,
        z, zcat, l, (l < NL - 1) ? 1 : 0);
  }

  // Graph readout
  zero_kernel<<<(NG * NL * HD + 255) / 256, 256, 0, stream>>>(gcat,
                                                              NG * NL * HD);
  pool_kernel<<<(NN * (NL * HD / 4) + 255) / 256, 256, 0, stream>>>(batch, zcat,
                                                                    gcat);
}